// SVDHead_44203803411127
// MI455X (gfx1250) — compile-verified
//
#include <hip/hip_runtime.h>
#include <hip/hip_bf16.h>
#include <math.h>

// Problem constants (from setup_inputs): B=16, D=128, N=2048, M=2048
#define BB   16
#define DD   128
#define NN   2048
#define MM   2048
#define EPSF 1e-8f

typedef __attribute__((ext_vector_type(2))) float v2f;
typedef __attribute__((ext_vector_type(8))) float v8f;

#define LDS_STRIDE 2052   // 2048 + 4 pad; row base stays 16B-aligned (2052*4 = 513*16)

// ---------------------------------------------------------------------------
// Kernel 1: fused  S = softmax(A^T B)  per batch, 16 rows per workgroup,
// full M in LDS. Emits v_wmma_f32_16x16x4_f32 (exact fp32, matches reference).
// Softmax passes are float4-vectorized: ds_load_b128 + global_store_b128 on
// the roofline-dominant 268 MB scores store.
// ---------------------------------------------------------------------------
__global__ __launch_bounds__(256)
void gemm_softmax_kernel(const float* __restrict__ srcE,   // (B, D, N)
                         const float* __restrict__ tgtE,   // (B, D, M)
                         float* __restrict__ scores,       // (B, N, M)
                         float* __restrict__ entW,         // (B, N)
                         int*   __restrict__ idxW)         // (B, N)
{
    __shared__ float lds_logits[16 * LDS_STRIDE];          // ~131 KB
    __shared__ float red_val[16][16];
    __shared__ int   red_idx[16][16];
    __shared__ float row_max[16];
    __shared__ float row_sum[16];
    __shared__ int   row_arg[16];

    const int tid   = threadIdx.x;
    const int wave  = tid >> 5;          // 0..7
    const int lane  = tid & 31;
    const int lo    = lane & 15;
    const int hi    = lane >> 4;         // half-wave select

    const int b      = blockIdx.x / (NN / 16);
    const int tile_n = blockIdx.x % (NN / 16);
    const int n0     = tile_n * 16;

    const float* Abase = srcE + (size_t)b * DD * NN;   // [d][n]
    const float* Bbase = tgtE + (size_t)b * DD * MM;   // [d][m]

    // ---- preload A fragments: A_w[16 x 4] per k-step, per ISA layout:
    // lanes 0-15: K = 4k+0, 4k+1 in .x/.y ; lanes 16-31: K = 4k+2, 4k+3
    v2f afrag[DD / 4];
#pragma unroll
    for (int k = 0; k < DD / 4; ++k) {
        const int d0 = 4 * k + 2 * hi;
        afrag[k].x = Abase[(size_t)d0 * NN + (n0 + lo)];
        afrag[k].y = Abase[(size_t)(d0 + 1) * NN + (n0 + lo)];
    }

    // ---- GEMM: each wave sweeps 16 of the 128 column tiles
    for (int t = wave; t < MM / 16; t += 8) {
        const int m0 = t * 16;
        v8f acc = {};
#pragma unroll
        for (int k = 0; k < DD / 4; ++k) {
            const int d0 = 4 * k + 2 * hi;
            v2f bfrag;
            bfrag.x = Bbase[(size_t)d0 * MM + (m0 + lo)];
            bfrag.y = Bbase[(size_t)(d0 + 1) * MM + (m0 + lo)];
            acc = __builtin_amdgcn_wmma_f32_16x16x4_f32(
                false, afrag[k], false, bfrag, (short)0, acc, false, false);
        }
        // C layout: VGPR j -> row j (+8 for upper half-wave), column = lo
#pragma unroll
        for (int j = 0; j < 8; ++j)
            lds_logits[(j + 8 * hi) * LDS_STRIDE + m0 + lo] = acc[j];
    }
    __syncthreads();

    // ---- softmax over M (workgroup-local, 16 threads per row, float4)
    const int row = tid >> 4;      // 0..15
    const int sub = tid & 15;      // 0..15
    const float* Lrow = &lds_logits[row * LDS_STRIDE];

    // pass 1: max + argmax (vectorized ds_load_b128)
    float mx = -3.402823466e38f; int mi = 0;
    for (int c0 = sub * 4; c0 < MM; c0 += 64) {
        const float4 v = *(const float4*)&Lrow[c0];
        if (v.x > mx) { mx = v.x; mi = c0; }
        if (v.y > mx) { mx = v.y; mi = c0 + 1; }
        if (v.z > mx) { mx = v.z; mi = c0 + 2; }
        if (v.w > mx) { mx = v.w; mi = c0 + 3; }
    }
    red_val[row][sub] = mx; red_idx[row][sub] = mi;
    __syncthreads();
    if (tid < 16) {
        float bm = red_val[tid][0]; int bi = red_idx[tid][0];
        for (int j = 1; j < 16; ++j) {
            float v = red_val[tid][j]; int i = red_idx[tid][j];
            if (v > bm || (v == bm && i < bi)) { bm = v; bi = i; }
        }
        row_max[tid] = bm; row_arg[tid] = bi;
    }
    __syncthreads();

    // pass 2: sum of exp
    const float m = row_max[row];
    float s = 0.f;
    for (int c0 = sub * 4; c0 < MM; c0 += 64) {
        const float4 v = *(const float4*)&Lrow[c0];
        s += __expf(v.x - m) + __expf(v.y - m) + __expf(v.z - m) + __expf(v.w - m);
    }
    red_val[row][sub] = s;
    __syncthreads();
    if (tid < 16) {
        float tot = 0.f;
        for (int j = 0; j < 16; ++j) tot += red_val[tid][j];
        row_sum[tid] = tot;
    }
    __syncthreads();

    // pass 3: entropy accumulate + normalized score store (global_store_b128)
    const float inv = 1.f / row_sum[row];
    float* Srow = scores + ((size_t)b * NN + (n0 + row)) * MM;
    float ent = 0.f;
    for (int c0 = sub * 4; c0 < MM; c0 += 64) {
        const float4 v = *(const float4*)&Lrow[c0];
        float4 p;
        p.x = __expf(v.x - m) * inv;
        p.y = __expf(v.y - m) * inv;
        p.z = __expf(v.z - m) * inv;
        p.w = __expf(v.w - m) * inv;
        ent += p.x * __logf(p.x + EPSF) + p.y * __logf(p.y + EPSF)
             + p.z * __logf(p.z + EPSF) + p.w * __logf(p.w + EPSF);
        *(float4*)&Srow[c0] = p;
    }
    red_val[row][sub] = ent;
    __syncthreads();
    if (tid < 16) {
        float tot = 0.f;
        for (int j = 0; j < 16; ++j) tot += red_val[tid][j];
        entW[(size_t)b * NN + n0 + tid] = tot;
        idxW[(size_t)b * NN + n0 + tid] = row_arg[tid];
    }
}

// ---------------------------------------------------------------------------
// Kernel 2: per-batch weighted Kabsch with 3x3 Jacobi SVD. One block / batch.
// ---------------------------------------------------------------------------
__device__ __forceinline__ float block_reduce_sum(float v, float* red) {
    const int tid = threadIdx.x;
    red[tid] = v;
    __syncthreads();
    for (int s = 128; s > 0; s >>= 1) {
        if (tid < s) red[tid] += red[tid + s];
        __syncthreads();
    }
    float r = red[0];
    __syncthreads();
    return r;
}

__device__ void jacobi3(float a[3][3], float v[3][3]) {
    for (int i = 0; i < 3; ++i)
        for (int j = 0; j < 3; ++j) v[i][j] = (i == j) ? 1.f : 0.f;
    for (int sweep = 0; sweep < 15; ++sweep) {
        for (int pq = 0; pq < 3; ++pq) {
            const int p = (pq == 2) ? 1 : 0;
            const int q = (pq == 0) ? 1 : 2;
            float apq = a[p][q];
            if (fabsf(apq) < 1e-30f) continue;
            float tau = (a[q][q] - a[p][p]) / (2.f * apq);
            float t   = (tau >= 0.f ? 1.f : -1.f) / (fabsf(tau) + sqrtf(1.f + tau * tau));
            float c   = 1.f / sqrtf(1.f + t * t);
            float sn  = t * c;
            for (int k = 0; k < 3; ++k) {   // columns
                float akp = a[k][p], akq = a[k][q];
                a[k][p] = c * akp - sn * akq;
                a[k][q] = sn * akp + c * akq;
            }
            for (int k = 0; k < 3; ++k) {   // rows
                float apk = a[p][k], aqk = a[q][k];
                a[p][k] = c * apk - sn * aqk;
                a[q][k] = sn * apk + c * aqk;
            }
            for (int k = 0; k < 3; ++k) {   // eigenvectors
                float vkp = v[k][p], vkq = v[k][q];
                v[k][p] = c * vkp - sn * vkq;
                v[k][q] = sn * vkp + c * vkq;
            }
        }
    }
}

__global__ __launch_bounds__(256)
void svd_head_kernel(const float* __restrict__ src,    // (B, 3, N)
                     const float* __restrict__ tgt,    // (B, 3, M)
                     const float* __restrict__ entW,   // (B, N)
                     const int*   __restrict__ idxW,   // (B, N)
                     float* __restrict__ R_out,        // (B, 3, 3)
                     float* __restrict__ t_out)        // (B, 3)
{
    __shared__ float red[256];
    __shared__ float sc_s[3], tc_s[3];

    const int b   = blockIdx.x;
    const int tid = threadIdx.x;
    const float* S  = src + (size_t)b * 3 * NN;
    const float* T  = tgt + (size_t)b * 3 * MM;
    const float* E  = entW + (size_t)b * NN;
    const int*   IX = idxW + (size_t)b * NN;

    // w = -1/(ent + eps), then normalize
    float wpart = 0.f;
    for (int n = tid; n < NN; n += 256) wpart += -1.f / (E[n] + EPSF);
    const float wsum = block_reduce_sum(wpart, red);
    const float wden = 1.f / (wsum + EPSF);

    // weighted centroids
    float cp[6] = {0.f, 0.f, 0.f, 0.f, 0.f, 0.f};
    for (int n = tid; n < NN; n += 256) {
        float wn = (-1.f / (E[n] + EPSF)) * wden;
        int j = IX[n];
        for (int i = 0; i < 3; ++i) {
            cp[i]     += S[i * NN + n] * wn;
            cp[3 + i] += T[i * MM + j] * wn;
        }
    }
    for (int i = 0; i < 3; ++i) {
        float r = block_reduce_sum(cp[i], red);
        if (tid == 0) sc_s[i] = r;
        r = block_reduce_sum(cp[3 + i], red);
        if (tid == 0) tc_s[i] = r;
    }
    __syncthreads();
    const float sc0 = sc_s[0], sc1 = sc_s[1], sc2 = sc_s[2];
    const float tc0 = tc_s[0], tc1 = tc_s[1], tc2 = tc_s[2];

    // H = sum_n (s_n - sc) ((corr_n - tc) * wn)^T
    float hp[9] = {0.f};
    for (int n = tid; n < NN; n += 256) {
        float wn = (-1.f / (E[n] + EPSF)) * wden;
        int j = IX[n];
        float sv[3] = { S[0 * NN + n] - sc0, S[1 * NN + n] - sc1, S[2 * NN + n] - sc2 };
        float cv[3] = { (T[0 * MM + j] - tc0) * wn,
                        (T[1 * MM + j] - tc1) * wn,
                        (T[2 * MM + j] - tc2) * wn };
        for (int i = 0; i < 3; ++i)
            for (int k = 0; k < 3; ++k) hp[3 * i + k] += sv[i] * cv[k];
    }
    float H[3][3];
    for (int i = 0; i < 9; ++i) {
        float r = block_reduce_sum(hp[i], red);
        H[i / 3][i % 3] = r;
    }

    if (tid == 0) {
        H[0][0] += EPSF; H[1][1] += EPSF; H[2][2] += EPSF;

        // eigen of H^T H -> V
        float A[3][3], V[3][3];
        for (int i = 0; i < 3; ++i)
            for (int j = 0; j < 3; ++j) {
                float s = 0.f;
                for (int k = 0; k < 3; ++k) s += H[k][i] * H[k][j];
                A[i][j] = s;
            }
        jacobi3(A, V);

        // sort eigenvalues descending (columns of V follow)
        float lam[3] = { A[0][0], A[1][1], A[2][2] };
        for (int i = 0; i < 2; ++i)
            for (int j = i + 1; j < 3; ++j)
                if (lam[j] > lam[i]) {
                    float tl = lam[i]; lam[i] = lam[j]; lam[j] = tl;
                    for (int k = 0; k < 3; ++k) {
                        float tv = V[k][i]; V[k][i] = V[k][j]; V[k][j] = tv;
                    }
                }

        // U columns: u_i = H v_i / sigma_i (cross-product fallback)
        float U[3][3];
        for (int i = 0; i < 3; ++i) {
            float u[3];
            for (int r = 0; r < 3; ++r)
                u[r] = H[r][0] * V[0][i] + H[r][1] * V[1][i] + H[r][2] * V[2][i];
            float nrm = sqrtf(u[0] * u[0] + u[1] * u[1] + u[2] * u[2]);
            if (nrm > 1e-20f) {
                float inv = 1.f / nrm;
                for (int r = 0; r < 3; ++r) U[r][i] = u[r] * inv;
            } else if (i == 2) {
                U[0][2] = U[1][0] * U[2][1] - U[2][0] * U[1][1];
                U[1][2] = U[2][0] * U[0][1] - U[0][0] * U[2][1];
                U[2][2] = U[0][0] * U[1][1] - U[1][0] * U[0][1];
            } else {
                for (int r = 0; r < 3; ++r) U[r][i] = (r == i) ? 1.f : 0.f;
            }
        }

        // r = V U^T ; det ; R = V diag(1,1,det) U^T
        float rV[3][3];
        for (int i = 0; i < 3; ++i)
            for (int j = 0; j < 3; ++j)
                rV[i][j] = V[i][0] * U[j][0] + V[i][1] * U[j][1] + V[i][2] * U[j][2];
        float det = rV[0][0] * (rV[1][1] * rV[2][2] - rV[1][2] * rV[2][1])
                  - rV[0][1] * (rV[1][0] * rV[2][2] - rV[1][2] * rV[2][0])
                  + rV[0][2] * (rV[1][0] * rV[2][1] - rV[1][1] * rV[2][0]);
        float R[3][3];
        for (int i = 0; i < 3; ++i)
            for (int j = 0; j < 3; ++j)
                R[i][j] = V[i][0] * U[j][0] + V[i][1] * U[j][1] + det * V[i][2] * U[j][2];

        float tc[3] = { tc0, tc1, tc2 };
        float sc[3] = { sc0, sc1, sc2 };
        for (int i = 0; i < 3; ++i) {
            float ti = tc[i] - (R[i][0] * sc[0] + R[i][1] * sc[1] + R[i][2] * sc[2]);
            t_out[b * 3 + i] = ti;
            for (int j = 0; j < 3; ++j) R_out[b * 9 + i * 3 + j] = R[i][j];
        }
    }
}

// ---------------------------------------------------------------------------
extern "C" void kernel_launch(void* const* d_in, const int* in_sizes, int n_in,
                              void* d_out, int out_size, void* d_ws, size_t ws_size,
                              hipStream_t stream) {
    const float* srcE = (const float*)d_in[0];   // (16,128,2048)
    const float* tgtE = (const float*)d_in[1];   // (16,128,2048)
    const float* src  = (const float*)d_in[2];   // (16,3,2048)
    const float* tgt  = (const float*)d_in[3];   // (16,3,2048)
    // d_in[4] = temperature: unused by the reference computation

    float* out      = (float*)d_out;
    float* R_out    = out;                       // 16*9
    float* t_out    = out + BB * 9;              // 16*3
    float* scoresO  = out + BB * 9 + BB * 3;     // 16*2048*2048

    float* entW = (float*)d_ws;                                  // 16*2048 f32
    int*   idxW = (int*)((char*)d_ws + (size_t)BB * NN * sizeof(float));

    gemm_softmax_kernel<<<dim3(BB * (NN / 16)), dim3(256), 0, stream>>>(
        srcE, tgtE, scoresO, entW, idxW);
    svd_head_kernel<<<dim3(BB), dim3(256), 0, stream>>>(
        src, tgt, entW, idxW, R_out, t_out);
}